// FeatureIntegration_71219147702622
// MI455X (gfx1250) — compile-verified
//
#include <hip/hip_runtime.h>
#include <hip/hip_bf16.h>
#include <cstdint>

typedef __attribute__((ext_vector_type(2))) float v2f;
typedef __attribute__((ext_vector_type(8))) float v8f;

// Fixed problem geometry from the reference (B=65536, D=1024):
constexpr int Dc  = 1024;      // feature dim / OC
constexpr int Gc  = 16384;     // B/4 groups
constexpr int ICc = 4096;      // 4*D
constexpr int CHUNKS = 32;     // row chunks for deterministic column stats

// ---------------------------------------------------------------------------
// Kernel 1: shared branch (0.5 * sum of 4 consecutive rows) -> out cols [0,D)
//           plus pids[::4] -> int region after the feature block
// ---------------------------------------------------------------------------
__global__ __launch_bounds__(256) void shared_pids_kernel(
    const float* __restrict__ shared, const int* __restrict__ pids,
    float* __restrict__ out, int* __restrict__ outp)
{
    size_t idx = (size_t)blockIdx.x * blockDim.x + threadIdx.x;
    size_t g = idx / Dc;
    size_t d = idx - g * Dc;
    const float* p = shared + (g * 4) * (size_t)Dc + d;
    float s = p[0] + p[Dc] + p[2 * Dc] + p[3 * Dc];
    out[g * (size_t)(2 * Dc) + d] = 0.5f * s;
    if (idx < (size_t)Gc) {
        outp[idx] = pids[idx * 4];
    }
}

// ---------------------------------------------------------------------------
// GEMM (NT): C[M,N] = A[M,K] * B[N,K]^T  using V_WMMA_F32_16X16X4_F32.
// One wave computes a 64x64 tile: 4x4 fragments of 16x16, K stepped by 4.
// All geometry is compile-time so fragment row offsets become load
// immediates (24-bit IOFFSET) and the loop keeps only 2 moving pointers.
// AFFINE=true applies per-K affine to A on the fly (folded BatchNorm).
//
// f32 WMMA fragment layout (wave32):
//   A 16x4 : lane l holds row M=l%16, K = kbase + 2*(l>=16) + {0,1}  (v2f)
//   B 4x16 : lane l holds col N=l%16, K = kbase + 2*(l>=16) + {0,1}  (v2f)
//   C 16x16: VGPR r holds M = r + 8*(l>=16), N = l%16                 (v8f)
// ---------------------------------------------------------------------------
template <int NTILES, int KTOT, int LDA, int LDB, int LDC, bool AFFINE>
__global__ __launch_bounds__(256) void gemm_nt_wmma_kernel(
    const float* __restrict__ A, const float* __restrict__ B,
    float* __restrict__ C,
    const float* __restrict__ kscale, const float* __restrict__ kshift)
{
    const int wave = threadIdx.x >> 5;
    const int lane = threadIdx.x & 31;
    const int half = lane >> 4;     // 0: K pair {0,1}, 1: K pair {2,3}
    const int lr   = lane & 15;

    const int tile = blockIdx.x * 8 + wave;
    const int m0 = (tile / NTILES) << 6;
    const int n0 = (tile % NTILES) << 6;

    v8f acc[4][4] = {};

    const float* Abase = A + (size_t)(m0 + lr) * LDA + 2 * half;
    const float* Bbase = B + (size_t)(n0 + lr) * LDB + 2 * half;
    const float* Sbase = AFFINE ? (kscale + 2 * half) : nullptr;
    const float* Hbase = AFFINE ? (kshift + 2 * half) : nullptr;

    for (int k = 0; k < KTOT; k += 4) {
        v2f sc, sh;
        if (AFFINE) {
            sc = *(const v2f*)(Sbase + k);
            sh = *(const v2f*)(Hbase + k);
        }
        v2f a[4], b[4];
#pragma unroll
        for (int i = 0; i < 4; ++i) {
            a[i] = *(const v2f*)(Abase + (size_t)(16 * i) * LDA + k);
            if (AFFINE) a[i] = a[i] * sc + sh;
        }
#pragma unroll
        for (int j = 0; j < 4; ++j) {
            b[j] = *(const v2f*)(Bbase + (size_t)(16 * j) * LDB + k);
        }
#pragma unroll
        for (int i = 0; i < 4; ++i) {
#pragma unroll
            for (int j = 0; j < 4; ++j) {
                acc[i][j] = __builtin_amdgcn_wmma_f32_16x16x4_f32(
                    /*neg_a=*/false, a[i], /*neg_b=*/false, b[j],
                    /*c_mod=*/(short)0, acc[i][j],
                    /*reuse_a=*/false, /*reuse_b=*/false);
            }
        }
    }

#pragma unroll
    for (int i = 0; i < 4; ++i) {
#pragma unroll
        for (int j = 0; j < 4; ++j) {
            const int col = n0 + 16 * j + lr;
#pragma unroll
            for (int r = 0; r < 8; ++r) {
                const int row = m0 + 16 * i + r + 8 * half;
                C[(size_t)row * LDC + col] = acc[i][j][r];
            }
        }
    }
}

// ---------------------------------------------------------------------------
// Column statistics, pass 1: per-chunk partial sum / sum-of-squares.
// grid = (D/256, CHUNKS), block = 256. Coalesced: adjacent lanes -> adjacent
// columns of the same row. No atomics => bit-deterministic.
// ---------------------------------------------------------------------------
template <int LD>
__global__ __launch_bounds__(256) void colstats_partial_kernel(
    const float* __restrict__ Y,
    float* __restrict__ psum, float* __restrict__ psq)
{
    const int col = blockIdx.x * blockDim.x + threadIdx.x;
    const int chunk = blockIdx.y;
    constexpr int ROWS = Gc / CHUNKS;
    const float* p = Y + (size_t)chunk * ROWS * LD + col;
    float s = 0.0f, s2 = 0.0f;
    for (int r = 0; r < ROWS; ++r) {
        float v = p[(size_t)r * LD];
        s += v;
        s2 += v * v;
    }
    psum[(size_t)chunk * Dc + col] = s;
    psq [(size_t)chunk * Dc + col] = s2;
}

// ---------------------------------------------------------------------------
// Column statistics, pass 2: reduce chunks in fixed order, fuse BN affine:
//   scale = gamma * rsqrt(var + eps),  shift = beta - mean * scale
// ---------------------------------------------------------------------------
__global__ __launch_bounds__(256) void colstats_finalize_kernel(
    const float* __restrict__ psum, const float* __restrict__ psq,
    const float* __restrict__ gamma, const float* __restrict__ beta,
    float* __restrict__ scale, float* __restrict__ shift)
{
    const int col = blockIdx.x * blockDim.x + threadIdx.x;
    float s = 0.0f, s2 = 0.0f;
#pragma unroll 4
    for (int c = 0; c < CHUNKS; ++c) {
        s  += psum[(size_t)c * Dc + col];
        s2 += psq [(size_t)c * Dc + col];
    }
    constexpr float invG = 1.0f / (float)Gc;
    const float mean = s * invG;
    const float var = s2 * invG - mean * mean;
    const float r = rsqrtf(var + 1e-5f);
    const float sc = gamma[col] * r;
    scale[col] = sc;
    shift[col] = beta[col] - mean * sc;
}

// ---------------------------------------------------------------------------
// Apply BN2 in place on the right half of the output block.
// ---------------------------------------------------------------------------
__global__ __launch_bounds__(256) void bn_apply_kernel(
    float* __restrict__ out, const float* __restrict__ scale,
    const float* __restrict__ shift)
{
    size_t idx = (size_t)blockIdx.x * blockDim.x + threadIdx.x;
    size_t g = idx / Dc;
    size_t c = idx - g * Dc;
    float* p = out + g * (size_t)(2 * Dc) + Dc + c;
    *p = (*p) * scale[c] + shift[c];
}

// ---------------------------------------------------------------------------
extern "C" void kernel_launch(void* const* d_in, const int* in_sizes, int n_in,
                              void* d_out, int out_size, void* d_ws, size_t ws_size,
                              hipStream_t stream)
{
    const float* shared   = (const float*)d_in[0];
    const float* specific = (const float*)d_in[1];
    const int*   pids     = (const int*)  d_in[2];
    const float* W1       = (const float*)d_in[3];
    const float* gamma1   = (const float*)d_in[4];
    const float* beta1    = (const float*)d_in[5];
    const float* W2       = (const float*)d_in[6];
    const float* gamma2   = (const float*)d_in[7];
    const float* beta2    = (const float*)d_in[8];

    float* out  = (float*)d_out;                         // [G, 2D] features
    int*   outp = (int*)d_out + (size_t)Gc * (2 * Dc);   // [G] pids

    // Workspace layout
    char* ws = (char*)d_ws;
    float* Y1     = (float*)ws;                                  // G*D floats
    float* psum   = (float*)(ws + (size_t)Gc * Dc * 4);          // CHUNKS*D
    float* psq    = psum + (size_t)CHUNKS * Dc;                  // CHUNKS*D
    float* scale1 = psq  + (size_t)CHUNKS * Dc;                  // D
    float* shift1 = scale1 + Dc;                                 // D
    float* scale2 = shift1 + Dc;                                 // D
    float* shift2 = scale2 + Dc;                                 // D

    const unsigned elemBlocks = (unsigned)(((size_t)Gc * Dc) / 256);
    const int tiles = (Gc / 64) * (Dc / 64);   // 4096, divisible by 8 waves

    // 1) shared branch + pids
    shared_pids_kernel<<<elemBlocks, 256, 0, stream>>>(shared, pids, out, outp);

    // 2) GEMM1: Y1[G,D] = specific[G,IC] @ W1[D,IC]^T
    gemm_nt_wmma_kernel<Dc / 64, ICc, ICc, ICc, Dc, false>
        <<<tiles / 8, 256, 0, stream>>>(specific, W1, Y1, nullptr, nullptr);

    // 3) BN1 stats over Y1 columns -> scale1/shift1
    {
        dim3 g1(Dc / 256, CHUNKS);
        colstats_partial_kernel<Dc><<<g1, 256, 0, stream>>>(Y1, psum, psq);
        colstats_finalize_kernel<<<Dc / 256, 256, 0, stream>>>(
            psum, psq, gamma1, beta1, scale1, shift1);
    }

    // 4) GEMM2 (BN1 folded into A-load): out[:, D:2D] = BN1(Y1) @ W2[D,D]^T
    gemm_nt_wmma_kernel<Dc / 64, Dc, Dc, Dc, 2 * Dc, true>
        <<<tiles / 8, 256, 0, stream>>>(Y1, W2, out + Dc, scale1, shift1);

    // 5) BN2 stats over out[:, D:2D] -> scale2/shift2, then apply in place
    {
        dim3 g1(Dc / 256, CHUNKS);
        colstats_partial_kernel<2 * Dc><<<g1, 256, 0, stream>>>(out + Dc, psum, psq);
        colstats_finalize_kernel<<<Dc / 256, 256, 0, stream>>>(
            psum, psq, gamma2, beta2, scale2, shift2);
        bn_apply_kernel<<<elemBlocks, 256, 0, stream>>>(out, scale2, shift2);
    }
}